// Model_12807592476809
// MI455X (gfx1250) — compile-verified
//
#include <hip/hip_runtime.h>
#include <hip/hip_bf16.h>

typedef __attribute__((ext_vector_type(2))) float v2f;
typedef __attribute__((ext_vector_type(8))) float v8f;

#define NHID   512
#define NLAY   4
#define NB     16
#define SEQ    511
#define LSEQ   512
#define NHEAD  8
#define NEDGE  131072
#define NNODE  (NB * SEQ)   /* 8176 */
#define MPAD   8192
#define QKVN   1536
#define FFN    2048

enum { EPI_BIAS = 0, EPI_BIAS_RELU = 1, EPI_BIAS_ADD = 2, EPI_ROWSCALE = 3 };

// ---- CDNA5 async global->LDS path (guarded; falls back to sync copies) ----
#if __has_builtin(__builtin_amdgcn_global_load_async_to_lds_b128) && \
    __has_builtin(__builtin_amdgcn_s_wait_asynccnt)
#define USE_ASYNC_LDS 1
#else
#define USE_ASYNC_LDS 0
#endif

typedef int v4i_vs __attribute__((vector_size(16)));
typedef __attribute__((address_space(1))) v4i_vs g_v4i;
typedef __attribute__((address_space(3))) v4i_vs l_v4i;

__device__ __forceinline__ void async_copy16(const float* gsrc, float* lds) {
#if USE_ASYNC_LDS
  __builtin_amdgcn_global_load_async_to_lds_b128(
      (g_v4i*)(uintptr_t)gsrc,
      (l_v4i*)(unsigned)(uintptr_t)lds,
      0, 0);
#else
  *(float4*)lds = *(const float4*)gsrc;
#endif
}
__device__ __forceinline__ void wait_async() {
#if USE_ASYNC_LDS
  __builtin_amdgcn_s_wait_asynccnt((short)0);
#endif
}

__device__ __forceinline__ v8f wmma4(v2f a, v2f b, v8f c) {
  return __builtin_amdgcn_wmma_f32_16x16x4_f32(false, a, false, b, (short)0, c,
                                               false, false);
}

// ---------------------------------------------------------------------------
// WMMA fp32 GEMM: C[M,N] = epi(A[M,K] @ W[K,N])
// Block tile 128x64, BK=32, double-buffered LDS (async A, transposed B).
// 8 waves; each wave owns a 32x32 C tile (4 accumulators).
// ---------------------------------------------------------------------------
__global__ __launch_bounds__(256) void gemm_wmma(
    const float* __restrict__ A, const float* __restrict__ W,
    const float* __restrict__ bias, const float* __restrict__ add,
    const float* __restrict__ rowscale, float addScale,
    float* __restrict__ C, int M, int N, int K, int epi) {
  __shared__ float As[2][128][34];   // row-major, 34-float stride: quads 8B-aligned in odd rows
  __shared__ float BsT[2][64][34];   // transposed: BsT[n][k]
  const int tid  = threadIdx.x;
  const int wave = tid >> 5, lane = tid & 31;
  const int half = lane >> 4, r = lane & 15;
  const int mt = (wave & 3) * 32;
  const int nt = (wave >> 2) * 32;
  const int rowBase = blockIdx.y * 128;
  const int colBase = blockIdx.x * 64;

  v8f acc00 = {}, acc01 = {}, acc10 = {}, acc11 = {};

  // staging: A 128x32 via async b128 (4 per thread); B 32x64 manual transpose
  const int am = tid >> 1;            // 0..127
  const int akb = (tid & 1) * 16;     // 0 or 16
  const int bkr = tid >> 3;           // 0..31
  const int bcb = (tid & 7) * 8;      // 0..56

#define STAGE_A(k0, buf)                                                    \
  {                                                                         \
    const float* src_ = A + (size_t)(rowBase + am) * K + (k0) + akb;        \
    async_copy16(src_, &As[buf][am][akb]);                                  \
    async_copy16(src_ + 4, &As[buf][am][akb + 4]);                          \
    async_copy16(src_ + 8, &As[buf][am][akb + 8]);                          \
    async_copy16(src_ + 12, &As[buf][am][akb + 12]);                        \
  }
#define STAGE_B(k0, buf)                                                    \
  {                                                                         \
    const float* src_ = W + (size_t)((k0) + bkr) * N + colBase + bcb;       \
    float t_[8];                                                            \
    _Pragma("unroll") for (int j = 0; j < 8; ++j) t_[j] = src_[j];          \
    _Pragma("unroll") for (int j = 0; j < 8; ++j)                           \
        BsT[buf][bcb + j][bkr] = t_[j];                                     \
  }

  STAGE_A(0, 0);
  STAGE_B(0, 0);
  wait_async();
  __syncthreads();

  int cur = 0;
  for (int k0 = 0; k0 < K; k0 += 32) {
    int nxt = cur ^ 1;
    if (k0 + 32 < K) {
      STAGE_A(k0 + 32, nxt);
      STAGE_B(k0 + 32, nxt);
    }
#pragma unroll
    for (int kk = 0; kk < 32; kk += 4) {
      int k = kk + 2 * half;
      v2f a0 = *(const v2f*)&As[cur][mt + r][k];
      v2f a1 = *(const v2f*)&As[cur][mt + 16 + r][k];
      v2f b0 = *(const v2f*)&BsT[cur][nt + r][k];
      v2f b1 = *(const v2f*)&BsT[cur][nt + 16 + r][k];
      acc00 = wmma4(a0, b0, acc00);
      acc01 = wmma4(a0, b1, acc01);
      acc10 = wmma4(a1, b0, acc10);
      acc11 = wmma4(a1, b1, acc11);
    }
    wait_async();
    __syncthreads();
    cur = nxt;
  }
#undef STAGE_A
#undef STAGE_B

  v8f accs[2][2] = {{acc00, acc01}, {acc10, acc11}};
#pragma unroll
  for (int im = 0; im < 2; ++im) {
#pragma unroll
    for (int i = 0; i < 8; ++i) {
      int row = rowBase + mt + im * 16 + i + 8 * half;
      if (row >= M) continue;
#pragma unroll
      for (int in = 0; in < 2; ++in) {
        int c = colBase + nt + in * 16 + r;
        float v = accs[im][in][i];
        if (epi == EPI_ROWSCALE) {
          v *= rowscale[row];
        } else {
          v += bias[c];
          if (epi == EPI_BIAS_RELU) v = fmaxf(v, 0.0f);
          else if (epi == EPI_BIAS_ADD) v += addScale * add[(size_t)row * N + c];
        }
        C[(size_t)row * N + c] = v;
      }
    }
  }
}

// ---------------------------------------------------------------------------
// Flash attention with WMMA. Grid (L/128, B*H). Block = 8 waves x 16 q-rows.
// Double-buffered K (async) / V-transposed (manual) chunks of 16 keys.
// ---------------------------------------------------------------------------
__global__ __launch_bounds__(256) void flash_attn(
    const float* __restrict__ qkv, float* __restrict__ O) {
  __shared__ float Ks[2][16][68];
  __shared__ float VsT[2][64][18];
  __shared__ float Pl[8][16][18];
  const int bh   = blockIdx.y;
  const int b    = bh >> 3, h = bh & 7;
  const int tid  = threadIdx.x;
  const int wave = tid >> 5, lane = tid & 31;
  const int half = lane >> 4, r = lane & 15;
  const int q0   = blockIdx.x * 128 + wave * 16;
  const float* base = qkv + (size_t)b * LSEQ * QKVN + h * 192;

  v2f qfrag[16];
  {
    const float* qrow = base + (size_t)(q0 + r) * QKVN;
#pragma unroll
    for (int s = 0; s < 16; ++s) {
      int k = 4 * s + 2 * half;
      qfrag[s].x = qrow[k] * 0.125f;       // fold 1/sqrt(64)
      qfrag[s].y = qrow[k + 1] * 0.125f;
    }
  }
  float mrow[8], lrow[8];
  v8f oacc[4] = {{}, {}, {}, {}};
#pragma unroll
  for (int i = 0; i < 8; ++i) { mrow[i] = -3.0e38f; lrow[i] = 0.0f; }

  const int srow = tid >> 4;          // 0..15
  const int scb  = (tid & 15) * 4;    // 0..60
#define STAGE_KV(j, buf)                                                     \
  {                                                                          \
    const float* kr_ = base + (size_t)((j) * 16 + srow) * QKVN + 64 + scb;   \
    async_copy16(kr_, &Ks[buf][srow][scb]);                                  \
    const float* vr_ = base + (size_t)((j) * 16 + srow) * QKVN + 128 + scb;  \
    float t0 = vr_[0], t1 = vr_[1], t2 = vr_[2], t3 = vr_[3];                \
    VsT[buf][scb + 0][srow] = t0;                                            \
    VsT[buf][scb + 1][srow] = t1;                                            \
    VsT[buf][scb + 2][srow] = t2;                                            \
    VsT[buf][scb + 3][srow] = t3;                                            \
  }

  STAGE_KV(0, 0);
  wait_async();
  __syncthreads();

  int cur = 0;
  for (int j = 0; j < 32; ++j) {
    int nxt = cur ^ 1;
    if (j + 1 < 32) STAGE_KV(j + 1, nxt);

    v8f s = {};
#pragma unroll
    for (int st = 0; st < 16; ++st) {
      v2f bb = *(const v2f*)&Ks[cur][r][4 * st + 2 * half];
      s = wmma4(qfrag[st], bb, s);
    }
#pragma unroll
    for (int i = 0; i < 8; ++i) {
      float sv = s[i];
      float rm = sv;
#pragma unroll
      for (int off = 1; off < 16; off <<= 1)
        rm = fmaxf(rm, __shfl_xor(rm, off, 16));
      float mnew = fmaxf(mrow[i], rm);
      float p    = __expf(sv - mnew);
      float rl   = p;
#pragma unroll
      for (int off = 1; off < 16; off <<= 1)
        rl += __shfl_xor(rl, off, 16);
      float alpha = __expf(mrow[i] - mnew);
      lrow[i] = lrow[i] * alpha + rl;
      mrow[i] = mnew;
      s[i]    = p;
      oacc[0][i] *= alpha; oacc[1][i] *= alpha;
      oacc[2][i] *= alpha; oacc[3][i] *= alpha;
    }
    // re-fragment P through per-wave LDS (DS ops are in-order within a wave)
#pragma unroll
    for (int i = 0; i < 8; ++i) Pl[wave][i + 8 * half][r] = s[i];
    __builtin_amdgcn_wave_barrier();
    v2f af[4];
#pragma unroll
    for (int ss = 0; ss < 4; ++ss)
      af[ss] = *(const v2f*)&Pl[wave][r][4 * ss + 2 * half];
#pragma unroll
    for (int ntile = 0; ntile < 4; ++ntile) {
#pragma unroll
      for (int ss = 0; ss < 4; ++ss) {
        v2f bb = *(const v2f*)&VsT[cur][ntile * 16 + r][4 * ss + 2 * half];
        oacc[ntile] = wmma4(af[ss], bb, oacc[ntile]);
      }
    }
    wait_async();
    __syncthreads();
    cur = nxt;
  }
#undef STAGE_KV

#pragma unroll
  for (int i = 0; i < 8; ++i) {
    float inv = 1.0f / lrow[i];
    int row = b * LSEQ + q0 + i + 8 * half;
#pragma unroll
    for (int ntile = 0; ntile < 4; ++ntile) {
      int col = h * 64 + ntile * 16 + r;
      O[(size_t)row * NHID + col] = oacc[ntile][i] * inv;
    }
  }
}

// ---------------------------------------------------------------------------
// Elementwise / graph kernels
// ---------------------------------------------------------------------------
__global__ void embed_kernel(const float* __restrict__ emb,
                             const float* __restrict__ pos,
                             const int* __restrict__ idxs,
                             float* __restrict__ H) {
  int gid = blockIdx.x * 256 + threadIdx.x;
  if (gid >= NB * LSEQ * NHID) return;
  int c = gid & (NHID - 1);
  int bl = gid >> 9;
  int l = bl & (LSEQ - 1);
  int b = bl >> 9;
  int tok = (l == 0) ? 0 : idxs[b * SEQ + (l - 1)];
  H[gid] = emb[(size_t)tok * NHID + c] + pos[(size_t)l * NHID + c];
}

__global__ void split_h(const float* __restrict__ H, float* __restrict__ xf,
                        float* __restrict__ zg) {
  int gid = blockIdx.x * 256 + threadIdx.x;
  if (gid >= NB * LSEQ * NHID) return;
  int c = gid & (NHID - 1);
  int bl = gid >> 9;
  int l = bl & (LSEQ - 1);
  int b = bl >> 9;
  float v = H[gid];
  if (l == 0) zg[b * NHID + c] = v;
  else xf[((size_t)(b * SEQ + l - 1)) * NHID + c] = v;
}

__global__ void deg_kernel(const int* __restrict__ idx, float* __restrict__ deg) {
  int e = blockIdx.x * 256 + threadIdx.x;
  if (e < NEDGE) atomicAdd(&deg[idx[e]], 1.0f);
}

__global__ void rs_kernel(const float* __restrict__ deg, float* __restrict__ rs,
                          int count) {
  int n = blockIdx.x * 256 + threadIdx.x;
  if (n < count) rs[n] = rsqrtf(fmaxf(deg[n], 1.0f));
}

__global__ void scatter_add(const float* __restrict__ x,
                            const int* __restrict__ src,
                            const int* __restrict__ dst,
                            float* __restrict__ agg, int F) {
  int e = blockIdx.x;
  int s = src[e], d = dst[e];
  const float* xs = x + (size_t)s * F;
  float* ad = agg + (size_t)d * F;
  for (int c = threadIdx.x; c < F; c += blockDim.x) atomicAdd(&ad[c], xs[c]);
}

__global__ void prescale(const float* __restrict__ agg,
                         const float* __restrict__ x,
                         const float* __restrict__ ind,
                         float* __restrict__ z) {
  int gid = blockIdx.x * 256 + threadIdx.x;
  if (gid >= NNODE * NHID) return;
  int n = gid >> 9;
  float s = 1.0f / (ind[n] + 1.0f);
  z[gid] = (agg[gid] + x[gid]) * s;
}

__global__ void combine_head(const float* __restrict__ agg64,
                             const float* __restrict__ rs_in,
                             const float* __restrict__ gcb,
                             float* __restrict__ xc, int head) {
  int gid = blockIdx.x * 256 + threadIdx.x;
  if (gid >= NNODE * 64) return;
  int c = gid & 63;
  int n = gid >> 6;
  xc[(size_t)n * NHID + head * 64 + c] = agg64[gid] * rs_in[n] + gcb[c];
}

// out[row] = add[row] + LayerNorm(x[row]) * g + b     (row length 512)
__global__ __launch_bounds__(256) void ln_add(
    const float* __restrict__ x, const float* __restrict__ add,
    const float* __restrict__ g, const float* __restrict__ bt,
    float* __restrict__ out) {
  int row = blockIdx.x;
  __shared__ float red[256];
  int t = threadIdx.x;
  const float* xr = x + (size_t)row * NHID;
  float a0 = xr[t], a1 = xr[t + 256];
  red[t] = a0 + a1;
  __syncthreads();
  for (int off = 128; off > 0; off >>= 1) {
    if (t < off) red[t] += red[t + off];
    __syncthreads();
  }
  float mu = red[0] * (1.0f / NHID);
  __syncthreads();
  float d0 = a0 - mu, d1 = a1 - mu;
  red[t] = d0 * d0 + d1 * d1;
  __syncthreads();
  for (int off = 128; off > 0; off >>= 1) {
    if (t < off) red[t] += red[t + off];
    __syncthreads();
  }
  float inv = rsqrtf(red[0] * (1.0f / NHID) + 1e-5f);
  const float* ar = add + (size_t)row * NHID;
  out[(size_t)row * NHID + t]       = ar[t]       + d0 * inv * g[t]       + bt[t];
  out[(size_t)row * NHID + t + 256] = ar[t + 256] + d1 * inv * g[t + 256] + bt[t + 256];
}

// ---------------------------------------------------------------------------
// Host side
// ---------------------------------------------------------------------------
static void gemm(const float* A, const float* W, const float* bias,
                 const float* add, const float* rowscale, float addScale,
                 float* C, int M, int N, int K, int epi, hipStream_t s) {
  dim3 grid(N / 64, (M + 127) / 128);
  gemm_wmma<<<grid, 256, 0, s>>>(A, W, bias, add, rowscale, addScale, C, M, N,
                                 K, epi);
}

extern "C" void kernel_launch(void* const* d_in, const int* in_sizes, int n_in,
                              void* d_out, int out_size, void* d_ws,
                              size_t ws_size, hipStream_t stream) {
  const float* emb   = (const float*)d_in[0];
  const float* pos   = (const float*)d_in[1];
  const float* qkv_w = (const float*)d_in[2];
  const float* qkv_b = (const float*)d_in[3];
  const float* ao_w  = (const float*)d_in[4];
  const float* ao_b  = (const float*)d_in[5];
  const float* ff1_w = (const float*)d_in[6];
  const float* ff1_b = (const float*)d_in[7];
  const float* ff2_w = (const float*)d_in[8];
  const float* ff2_b = (const float*)d_in[9];
  const float* sg1_w = (const float*)d_in[10];
  const float* sg1_b = (const float*)d_in[11];
  const float* sg2_w = (const float*)d_in[12];
  const float* sg2_b = (const float*)d_in[13];
  const float* gc3_w = (const float*)d_in[14];
  const float* gc3_b = (const float*)d_in[15];
  const float* gf1_w = (const float*)d_in[16];
  const float* gf1_b = (const float*)d_in[17];
  const float* gf2_w = (const float*)d_in[18];
  const float* gf2_b = (const float*)d_in[19];
  const float* ln_g  = (const float*)d_in[20];
  const float* ln_b  = (const float*)d_in[21];
  const int*   idxs  = (const int*)d_in[22];
  // d_in[23] = mask (all true -> unused)
  const int* gt_src   = (const int*)d_in[24];
  const int* gt_dst   = (const int*)d_in[25];
  const int* attr_src = (const int*)d_in[26];
  const int* attr_dst = (const int*)d_in[27];

  float* out = (float*)d_out;
  float* zbar = out;                                // 8176 x 512
  float* zg   = out + (size_t)NNODE * NHID;         // 16 x 512

  float* ws = (float*)d_ws;
  const size_t ROW = (size_t)MPAD * NHID;  // 4M floats = 16MB
  float* H   = ws;
  float* BIG = H + ROW;            // 4*ROW: qkv / ff-hidden / sage slots
  float* OB  = BIG + 4 * ROW;      // attn out; aliased: AGG
  float* XB  = OB + ROW;           // x;        aliased: Z
  float* XF  = XB + ROW;           // hg flattened
  float* X2  = XF + ROW;
  float* FO  = X2 + ROW;
  float* F64 = FO + ROW;           // 8192*64
  float* A64 = F64 + (size_t)MPAD * 64;
  float* DEG = A64 + (size_t)MPAD * 64;  // 8 arrays of 8192
  size_t needed = (size_t)(DEG + 8 * MPAD - ws) * sizeof(float);
  if (ws_size < needed) return;

  float* AGG = OB;
  float* Z   = XB;
  float* S1  = BIG;
  float* S2  = BIG + ROW;
  float* XC  = BIG + 2 * ROW;
  float* FH  = BIG + 3 * ROW;

  const int TOT = NB * LSEQ * NHID;

  // ---- embedding ----
  embed_kernel<<<(TOT + 255) / 256, 256, 0, stream>>>(emb, pos, idxs, H);

  // ---- transformer layers ----
  for (int l = 0; l < NLAY; ++l) {
    gemm(H, qkv_w + (size_t)l * NHID * QKVN, qkv_b + (size_t)l * QKVN, nullptr,
         nullptr, 0.0f, BIG, MPAD, QKVN, NHID, EPI_BIAS, stream);
    flash_attn<<<dim3(4, NB * NHEAD), 256, 0, stream>>>(BIG, OB);
    // x = 2h + o @ Wo + bo
    gemm(OB, ao_w + (size_t)l * NHID * NHID, ao_b + (size_t)l * NHID, H,
         nullptr, 2.0f, XB, MPAD, NHID, NHID, EPI_BIAS_ADD, stream);
    // t = relu(x @ W1 + b1)
    gemm(XB, ff1_w + (size_t)l * NHID * FFN, ff1_b + (size_t)l * FFN, nullptr,
         nullptr, 0.0f, BIG, MPAD, FFN, NHID, EPI_BIAS_RELU, stream);
    // h = x + t @ W2 + b2
    gemm(BIG, ff2_w + (size_t)l * FFN * NHID, ff2_b + (size_t)l * NHID, XB,
         nullptr, 1.0f, H, MPAD, NHID, FFN, EPI_BIAS_ADD, stream);
  }

  // ---- split zg / xf ----
  split_h<<<(TOT + 255) / 256, 256, 0, stream>>>(H, XF, zg);

  // ---- degrees + inverse-sqrt scales ----
  (void)hipMemsetAsync(DEG, 0, 8 * MPAD * sizeof(float), stream);
  deg_kernel<<<NEDGE / 256, 256, 0, stream>>>(gt_dst, DEG + 0 * MPAD);
  deg_kernel<<<NEDGE / 256, 256, 0, stream>>>(gt_src, DEG + 1 * MPAD);
  deg_kernel<<<NEDGE / 256, 256, 0, stream>>>(attr_dst, DEG + 2 * MPAD);
  deg_kernel<<<NEDGE / 256, 256, 0, stream>>>(attr_src, DEG + 3 * MPAD);
  rs_kernel<<<(4 * MPAD + 255) / 256, 256, 0, stream>>>(DEG, DEG + 4 * MPAD,
                                                        4 * MPAD);

  static const int PATTERN[8] = {0, 0, 1, 1, 0, 0, 1, 1};
  const int ETOT = NNODE * NHID;

  for (int i = 0; i < NHEAD; ++i) {
    int g = PATTERN[i];
    const int* src = g ? attr_src : gt_src;
    const int* dst = g ? attr_dst : gt_dst;
    const float* ind    = DEG + (g ? 2 : 0) * MPAD;
    const float* rs_in  = DEG + (4 + (g ? 2 : 0)) * MPAD;
    const float* rs_out = DEG + (5 + (g ? 2 : 0)) * MPAD;

    // sage1
    (void)hipMemsetAsync(AGG, 0, ROW * sizeof(float), stream);
    scatter_add<<<NEDGE, 256, 0, stream>>>(XF, src, dst, AGG, NHID);
    prescale<<<(ETOT + 255) / 256, 256, 0, stream>>>(AGG, XF, ind, Z);
    gemm(Z, sg1_w + (size_t)i * NHID * NHID, sg1_b + (size_t)i * NHID, nullptr,
         nullptr, 0.0f, S1, NNODE, NHID, NHID, EPI_BIAS, stream);
    // sage2
    (void)hipMemsetAsync(AGG, 0, ROW * sizeof(float), stream);
    scatter_add<<<NEDGE, 256, 0, stream>>>(S1, src, dst, AGG, NHID);
    prescale<<<(ETOT + 255) / 256, 256, 0, stream>>>(AGG, S1, ind, Z);
    gemm(Z, sg2_w + (size_t)i * NHID * NHID, sg2_b + (size_t)i * NHID, nullptr,
         nullptr, 0.0f, S2, NNODE, NHID, NHID, EPI_BIAS, stream);
    // gc3: f = (o @ W3) * rs_out[row]
    gemm(S2, gc3_w + (size_t)i * NHID * 64, nullptr, nullptr, rs_out, 0.0f,
         F64, NNODE, 64, NHID, EPI_ROWSCALE, stream);
    (void)hipMemsetAsync(A64, 0, (size_t)MPAD * 64 * sizeof(float), stream);
    scatter_add<<<NEDGE, 64, 0, stream>>>(F64, src, dst, A64, 64);
    combine_head<<<(NNODE * 64 + 255) / 256, 256, 0, stream>>>(
        A64, rs_in, gc3_b + (size_t)i * 64, XC, i);
  }

  // ---- x2 = hg + LN(xc) ----
  ln_add<<<NNODE, 256, 0, stream>>>(XC, XF, ln_g, ln_b, X2);
  // ---- gcn FFN ----
  gemm(X2, gf1_w, gf1_b, nullptr, nullptr, 0.0f, FH, NNODE, NHID, NHID,
       EPI_BIAS_RELU, stream);
  gemm(FH, gf2_w, gf2_b, nullptr, nullptr, 0.0f, FO, NNODE, NHID, NHID,
       EPI_BIAS, stream);
  // ---- zbar = x2 + LN(ff) ----
  ln_add<<<NNODE, 256, 0, stream>>>(FO, X2, ln_g, ln_b, zbar);
}